// SelfAttentionNaive_16449724744765
// MI455X (gfx1250) — compile-verified
//
#include <hip/hip_runtime.h>
#include <hip/hip_bf16.h>
#include <stdint.h>

typedef __attribute__((ext_vector_type(2))) float v2f;
typedef __attribute__((ext_vector_type(4))) float v4f;
typedef __attribute__((ext_vector_type(8))) float v8f;

#define NFEAT 512
#define NHID  64
#define NPIX  4096
#define NBATCH 8

// ---------------------------------------------------------------------------
// Kernel A: fused projections.  y[o,n] = sum_c W[o,c] * x[b,c,n],  o in [0,640)
//   o 0..63   -> fx (wf),  o 64..127 -> gx (wg),  o 128..639 -> hx (wh)
// One 16x16 C tile per wave; K stepped by 4 with V_WMMA_F32_16X16X4_F32.
// A layout (16x4 f32): lane<16 holds {k,k+1}, lane>=16 holds {k+2,k+3}, row = lane&15.
// B layout (4x16 f32): same k split, col = lane&15.
// ---------------------------------------------------------------------------
__global__ __launch_bounds__(256)
void proj_kernel(const float* __restrict__ x,
                 const float* __restrict__ wf,
                 const float* __restrict__ wg,
                 const float* __restrict__ wh,
                 float* __restrict__ fx, float* __restrict__ gx,
                 float* __restrict__ hx)
{
    const int b    = blockIdx.z;
    const int o0   = blockIdx.y * 16;
    const int lane = threadIdx.x & 31;
    const int wave = threadIdx.x >> 5;                 // wave32
    const int n0   = blockIdx.x * 128 + wave * 16;

    const float* xb = x + (size_t)b * NFEAT * NPIX;

    const float* W;
    float* dst;
    int orow = o0;
    if (o0 < 64)        { W = wf; dst = fx + (size_t)b * NHID  * NPIX; }
    else if (o0 < 128)  { W = wg; dst = gx + (size_t)b * NHID  * NPIX; orow = o0 - 64; }
    else                { W = wh; dst = hx + (size_t)b * NFEAT * NPIX; orow = o0 - 128; }

    const int l15   = lane & 15;
    const int khalf = (lane >> 4) << 1;                // 0 or 2
    const int ncol  = n0 + l15;

    v8f acc = {};
#pragma unroll 4
    for (int k = 0; k < NFEAT; k += 4) {
        const int kk = k + khalf;
        v2f a = *(const v2f*)(W + (size_t)(orow + l15) * NFEAT + kk);   // b64
        v2f bv;
        bv.x = xb[(size_t)kk * NPIX + ncol];
        bv.y = xb[(size_t)(kk + 1) * NPIX + ncol];
        acc = __builtin_amdgcn_wmma_f32_16x16x4_f32(false, a, false, bv,
                                                    (short)0, acc, false, false);
    }

    // C layout: row-in-tile = j + 8*(lane>=16), col = lane&15
    const int rbase = (lane >> 4) << 3;
#pragma unroll
    for (int j = 0; j < 8; ++j)
        dst[(size_t)(orow + rbase + j) * NPIX + ncol] = acc[j];
}

// ---------------------------------------------------------------------------
// Kernel B: flash attention.  One (batch, 32-column query tile) per block.
//   Q = gx[:, m0:m0+32]  (64 x 32, LDS, loaded once)
//   loop over key blocks n0 (step 32), K tile double-buffered via
//   GLOBAL_LOAD_ASYNC_TO_LDS_B128 (ASYNCcnt):
//     S[n,m] = sum_c fx[c,n]*gx[c,m]   (waves 0-3, WMMA, C spilled to LDS)
//     online softmax over the KEY axis n (== reference softmax over dim=1)
//     acc[c,m] += hx[c, n-block] * P   (all 8 waves, 64 channels each, WMMA)
//   out = gamma * acc/L + x
// ---------------------------------------------------------------------------
__global__ __launch_bounds__(256)
void flash_kernel(const float* __restrict__ fx, const float* __restrict__ gx,
                  const float* __restrict__ hx, const float* __restrict__ x,
                  const float* __restrict__ gamma, float* __restrict__ out)
{
    __shared__ float Qs[64][32];
    __shared__ float Ks[2][64][32];          // double-buffered K tile
    __shared__ float Ps[32][32];
    __shared__ float Mrun[32], Lrun[32], Scale[32];

    const int b    = blockIdx.y;
    const int m0   = blockIdx.x * 32;
    const int tid  = threadIdx.x;
    const int lane = tid & 31;
    const int wave = tid >> 5;
    const int c0   = wave * 64;            // this wave's 64-channel V slice

    const float* fxb = fx + (size_t)b * NHID  * NPIX;
    const float* gxb = gx + (size_t)b * NHID  * NPIX;
    const float* hxb = hx + (size_t)b * NFEAT * NPIX;

    // ---- load Q tile (64x32): 8 floats per thread, coalesced ----
    {
        const int r = tid >> 2;
        const int c = (tid & 3) * 8;
        const float* src = gxb + (size_t)r * NPIX + m0 + c;
        *(v4f*)(&Qs[r][c])     = *(const v4f*)(src);
        *(v4f*)(&Qs[r][c + 4]) = *(const v4f*)(src + 4);
    }
    if (tid < 32) { Mrun[tid] = -3.0e38f; Lrun[tid] = 0.0f; }

    // async fill of K tile: each thread moves 32B (2 x b128), LDS addr is the
    // low 32 bits of the generic shared pointer (flat LDS aperture rule).
    const int kr = tid >> 2;
    const int kc = (tid & 3) * 8;
    auto issueK = [&](int n0, int bufIdx) {
        const float* src = fxb + (size_t)kr * NPIX + n0 + kc;
        const uint32_t lds0 = (uint32_t)(uintptr_t)(&Ks[bufIdx][kr][kc]);
        asm volatile(
            "global_load_async_to_lds_b128 %0, %1, off\n\t"
            "global_load_async_to_lds_b128 %0, %1, off offset:16"
            :: "v"(lds0), "v"(src) : "memory");
    };

    issueK(0, 0);                            // prime tile 0

    const v8f zacc = {};
    v8f acc[4][2];
#pragma unroll
    for (int rt = 0; rt < 4; ++rt) { acc[rt][0] = zacc; acc[rt][1] = zacc; }

    const int l15   = lane & 15;
    const int khalf = (lane >> 4) << 1;

    for (int it = 0; it < NPIX / 32; ++it) {
        const int n0  = it * 32;
        const int buf = it & 1;

        asm volatile("s_wait_asynccnt 0x0" ::: "memory");   // tile `it` landed
        __syncthreads();

        if (it + 1 < NPIX / 32)                             // overlap next fill
            issueK(n0 + 32, buf ^ 1);

        // ---- S tile: waves 0-3 each compute one 16x16 subtile ----
        if (wave < 4) {
            const int rn = (wave >> 1) * 16;          // key-row offset
            const int cm = (wave & 1) * 16;           // query-col offset
            v8f s = zacc;
#pragma unroll
            for (int k = 0; k < NHID; k += 4) {
                const int kk = k + khalf;
                v2f a, bq;
                a.x  = Ks[buf][kk][rn + l15];  a.y  = Ks[buf][kk + 1][rn + l15];
                bq.x = Qs[kk][cm + l15];       bq.y = Qs[kk + 1][cm + l15];
                s = __builtin_amdgcn_wmma_f32_16x16x4_f32(false, a, false, bq,
                                                          (short)0, s, false, false);
            }
            const int rbase = rn + ((lane >> 4) << 3);
#pragma unroll
            for (int j = 0; j < 8; ++j)
                Ps[rbase + j][cm + l15] = s[j];
        }
        __syncthreads();

        // ---- online softmax over key axis, one thread per query column ----
        if (tid < 32) {
            float bm = Ps[0][tid];
#pragma unroll
            for (int n = 1; n < 32; ++n) bm = fmaxf(bm, Ps[n][tid]);
            const float mOld = Mrun[tid];
            const float mNew = fmaxf(mOld, bm);
            const float sc   = __expf(mOld - mNew);
            float sum = 0.0f;
#pragma unroll
            for (int n = 0; n < 32; ++n) {
                const float p = __expf(Ps[n][tid] - mNew);
                Ps[n][tid] = p;
                sum += p;
            }
            Mrun[tid]  = mNew;
            Lrun[tid]  = Lrun[tid] * sc + sum;
            Scale[tid] = sc;
        }
        __syncthreads();

        // ---- rescale accumulators, then acc += hx_tile * P ----
        const float sc0 = Scale[l15];
        const float sc1 = Scale[16 + l15];
#pragma unroll
        for (int rt = 0; rt < 4; ++rt) {
#pragma unroll
            for (int j = 0; j < 8; ++j) { acc[rt][0][j] *= sc0; acc[rt][1][j] *= sc1; }
        }
#pragma unroll
        for (int k = 0; k < 32; k += 4) {
            const int kk = k + khalf;
            v2f b0, b1;
            b0.x = Ps[kk][l15];       b0.y = Ps[kk + 1][l15];
            b1.x = Ps[kk][16 + l15];  b1.y = Ps[kk + 1][16 + l15];
#pragma unroll
            for (int rt = 0; rt < 4; ++rt) {
                const int crow = c0 + rt * 16 + l15;
                v2f a = *(const v2f*)(hxb + (size_t)crow * NPIX + n0 + kk);  // b64
                acc[rt][0] = __builtin_amdgcn_wmma_f32_16x16x4_f32(false, a, false, b0,
                                                                   (short)0, acc[rt][0], false, false);
                acc[rt][1] = __builtin_amdgcn_wmma_f32_16x16x4_f32(false, a, false, b1,
                                                                   (short)0, acc[rt][1], false, false);
            }
        }
    }

    // ---- epilogue: out = gamma * acc/L + x ----
    const float g    = gamma[0];
    const float inv0 = 1.0f / Lrun[l15];
    const float inv1 = 1.0f / Lrun[16 + l15];
    const float* xb  = x   + (size_t)b * NFEAT * NPIX;
    float*       ob  = out + (size_t)b * NFEAT * NPIX;
    const int rbase  = (lane >> 4) << 3;
#pragma unroll
    for (int rt = 0; rt < 4; ++rt) {
#pragma unroll
        for (int ct = 0; ct < 2; ++ct) {
            const float inv = ct ? inv1 : inv0;
            const int mcol  = m0 + ct * 16 + l15;
#pragma unroll
            for (int j = 0; j < 8; ++j) {
                const int crow = c0 + rt * 16 + rbase + j;
                const size_t idx = (size_t)crow * NPIX + mcol;
                ob[idx] = g * (acc[rt][ct][j] * inv) + xb[idx];
            }
        }
    }
}

// ---------------------------------------------------------------------------
extern "C" void kernel_launch(void* const* d_in, const int* in_sizes, int n_in,
                              void* d_out, int out_size, void* d_ws, size_t ws_size,
                              hipStream_t stream)
{
    const float* x     = (const float*)d_in[0];
    const float* wf    = (const float*)d_in[1];
    const float* wg    = (const float*)d_in[2];
    const float* wh    = (const float*)d_in[3];
    const float* gamma = (const float*)d_in[4];
    float* out = (float*)d_out;

    // workspace: fx (8.4MB) | gx (8.4MB) | hx (67MB)
    float* ws = (float*)d_ws;
    float* fx = ws;
    float* gx = fx + (size_t)NBATCH * NHID * NPIX;
    float* hx = gx + (size_t)NBATCH * NHID * NPIX;

    dim3 gA(NPIX / 128, (2 * NHID + NFEAT) / 16, NBATCH);   // 32 x 40 x 8
    proj_kernel<<<gA, 256, 0, stream>>>(x, wf, wg, wh, fx, gx, hx);

    dim3 gB(NPIX / 32, NBATCH);                              // 128 x 8
    flash_kernel<<<gB, 256, 0, stream>>>(fx, gx, hx, x, gamma, out);
}